// Vanilla_Quantizer_17995912970290
// MI455X (gfx1250) — compile-verified
//
#include <hip/hip_runtime.h>
#include <hip/hip_bf16.h>

typedef __attribute__((ext_vector_type(2))) float v2f;
typedef __attribute__((ext_vector_type(8))) float v8f;

#define NTOT   16384   // B*H*W rows
#define DDIM   256
#define KTOT   8192
#define KSPLIT 2
#define CB_PITCH 17    // LDS pitch (floats) for 16-col codebook tile -> conflict-free

// ---------------- K1: transpose z [B,C,HW] -> zt [B*HW, C] ----------------
__global__ __launch_bounds__(256) void k_transpose_z(const float* __restrict__ z,
                                                     float* __restrict__ zt) {
  __shared__ float tile[32][33];
  const int b = blockIdx.z, hw0 = blockIdx.x * 32, c0 = blockIdx.y * 32;
  const float* src = z + (size_t)b * DDIM * 1024;
#pragma unroll
  for (int i = 0; i < 4; ++i) {
    int cl = threadIdx.y + 8 * i;
    tile[cl][threadIdx.x] = src[(size_t)(c0 + cl) * 1024 + hw0 + threadIdx.x];
  }
  __syncthreads();
  float* dst = zt + (size_t)b * 1024 * DDIM;
#pragma unroll
  for (int i = 0; i < 4; ++i) {
    int hwr = threadIdx.y + 8 * i;
    dst[(size_t)(hw0 + hwr) * DDIM + c0 + threadIdx.x] = tile[threadIdx.x][hwr];
  }
}

// ---------------- K2: cnorm[k] = ||codebook[k]||^2 (one wave per row) -----
__global__ __launch_bounds__(256) void k_cnorm(const float* __restrict__ cb,
                                               float* __restrict__ cnorm) {
  const int wave = threadIdx.x >> 5, lane = threadIdx.x & 31;
  const int k = blockIdx.x * 8 + wave;
  const float* row = cb + (size_t)k * DDIM + lane * 8;
  float s = 0.f;
#pragma unroll
  for (int i = 0; i < 8; ++i) { float v = row[i]; s += v * v; }
#pragma unroll
  for (int off = 16; off >= 1; off >>= 1) s += __shfl_xor(s, off, 32);
  if (lane == 0) cnorm[k] = s;
}

// ---------------- K3: WMMA GEMM + running argmin --------------------------
// Wave owns 16 z-rows; A (16x256 f32) resident in 128 VGPRs for whole kernel.
// WG = 8 waves = 128 rows; blockIdx.x in [0,128); blockIdx.y = K half.
// Codebook tiles are double-buffered through registers: next tile's global
// loads are issued right after committing the current tile to LDS, so HBM/L2
// latency overlaps the 64-WMMA compute phase.
__global__ __launch_bounds__(256) void k_gemm_argmin(
    const float* __restrict__ zt, const float* __restrict__ cb,
    const float* __restrict__ cnorm, float* __restrict__ minval,
    int* __restrict__ minidx) {
  __shared__ float cbT[DDIM * CB_PITCH];  // [d][col], 17-float pitch

  const int lane = threadIdx.x & 31;
  const int half = lane >> 4;      // 0: K={0,1} of chunk, 1: K={2,3}
  const int l15  = lane & 15;
  const int wave = threadIdx.x >> 5;
  const int n0   = blockIdx.x * 128 + wave * 16;

  // A fragments (ISA 32-bit A 16x4 layout): a[j].x = zt[row][4j+2*half],
  // a[j].y = zt[row][4j+1+2*half]; row = n0 + (lane&15).
  v2f a[64];
  {
    const float* zrow = zt + (size_t)(n0 + l15) * DDIM + 2 * half;
#pragma unroll
    for (int j = 0; j < 64; ++j) a[j] = *(const v2f*)(zrow + 4 * j);
  }

  float minv[8];
  int   mini[8];
#pragma unroll
  for (int i = 0; i < 8; ++i) { minv[i] = 3.402823466e38f; mini[i] = 0; }

  const int ITERS = (KTOT / KSPLIT) / 16;
  int k0 = blockIdx.y * (KTOT / KSPLIT);

  // Prefetch first tile: thread t owns d = t; coalesced along d per column.
  float pre[16];
#pragma unroll
  for (int col = 0; col < 16; ++col)
    pre[col] = cb[(size_t)(k0 + col) * DDIM + threadIdx.x];

  for (int it = 0; it < ITERS; ++it, k0 += 16) {
    __syncthreads();  // previous iteration's cbT reads done
    // Commit prefetched tile transposed: cbT[d][col]; LDS writes at
    // (17*t + col) hit 32 distinct banks (gcd(17,64)=1).
#pragma unroll
    for (int col = 0; col < 16; ++col)
      cbT[threadIdx.x * CB_PITCH + col] = pre[col];
    __syncthreads();

    // Issue next tile's global loads; latency overlaps the WMMA phase.
    if (it + 1 < ITERS) {
      const int kn = k0 + 16;
#pragma unroll
      for (int col = 0; col < 16; ++col)
        pre[col] = cb[(size_t)(kn + col) * DDIM + threadIdx.x];
    }

    const float cn = cnorm[k0 + l15];

    v8f c = {0.f, 0.f, 0.f, 0.f, 0.f, 0.f, 0.f, 0.f};
#pragma unroll
    for (int j = 0; j < 64; ++j) {
      // B 4x16 fragment, mirror of A layout across half-waves.
      v2f b;
      b.x = cbT[(4 * j + 2 * half) * CB_PITCH + l15];
      b.y = cbT[(4 * j + 1 + 2 * half) * CB_PITCH + l15];
      c = __builtin_amdgcn_wmma_f32_16x16x4_f32(
          false, a[j], false, b, (short)0, c, false, false);
    }

    // C layout: lane -> column (k0 + l15), VGPR i -> row i + 8*half.
    const int colk = k0 + l15;
#pragma unroll
    for (int i = 0; i < 8; ++i) {
      float s = cn - 2.0f * c[i];
      bool better = s < minv[i];
      mini[i] = better ? colk : mini[i];
      minv[i] = better ? s : minv[i];
    }
  }

  // Reduce min across the 16 lanes of each half (halves hold disjoint rows).
#pragma unroll
  for (int i = 0; i < 8; ++i) {
    float v = minv[i]; int id = mini[i];
#pragma unroll
    for (int off = 8; off >= 1; off >>= 1) {
      float ov = __shfl_xor(v, off, 32);
      int   oi = __shfl_xor(id, off, 32);
      if (ov < v || (ov == v && oi < id)) { v = ov; id = oi; }
    }
    if (l15 == 0) {
      int row = n0 + i + 8 * half;
      minval[blockIdx.y * NTOT + row] = v;
      minidx[blockIdx.y * NTOT + row] = id;
    }
  }
}

// ---------------- K3b: combine the KSPLIT partial argmins -----------------
__global__ __launch_bounds__(256) void k_combine(const float* __restrict__ minval,
                                                 const int* __restrict__ minidx,
                                                 int* __restrict__ token) {
  int n = blockIdx.x * 256 + threadIdx.x;
  float v0 = minval[n], v1 = minval[NTOT + n];
  int   i0 = minidx[n], i1 = minidx[NTOT + n];
  token[n] = (v1 < v0 || (v1 == v0 && i1 < i0)) ? i1 : i0;
}

// ---------------- K4: residual sumsq + histogram --------------------------
__global__ __launch_bounds__(256) void k_stats(const float* __restrict__ zt,
                                               const float* __restrict__ cb,
                                               const int* __restrict__ token,
                                               float* __restrict__ total,
                                               int* __restrict__ hist) {
  __shared__ float red[8];
  const int n = blockIdx.x;
  const int t = token[n];
  float q  = cb[(size_t)t * DDIM + threadIdx.x];
  float zv = zt[(size_t)n * DDIM + threadIdx.x];
  float r = q - zv;
  float s = r * r;
#pragma unroll
  for (int off = 16; off >= 1; off >>= 1) s += __shfl_xor(s, off, 32);
  if ((threadIdx.x & 31) == 0) red[threadIdx.x >> 5] = s;
  __syncthreads();
  if (threadIdx.x == 0) {
    float tot = 0.f;
#pragma unroll
    for (int w = 0; w < 8; ++w) tot += red[w];
    atomicAdd(total, tot);
    atomicAdd(&hist[t], 1);
  }
}

// ---------------- K5: out[b][c][hw] = codebook[token[n]][c] ---------------
__global__ __launch_bounds__(256) void k_write_out(const float* __restrict__ cb,
                                                   const int* __restrict__ token,
                                                   float* __restrict__ out) {
  __shared__ float tile[32][33];
  __shared__ int tok[32];
  const int b = blockIdx.z, hw0 = blockIdx.x * 32, c0 = blockIdx.y * 32;
  if (threadIdx.y == 0) tok[threadIdx.x] = token[b * 1024 + hw0 + threadIdx.x];
  __syncthreads();
#pragma unroll
  for (int i = 0; i < 4; ++i) {
    int hw = threadIdx.y + 8 * i;
    tile[hw][threadIdx.x] = cb[(size_t)tok[hw] * DDIM + c0 + threadIdx.x];
  }
  __syncthreads();
  float* dst = out + (size_t)b * DDIM * 1024;
#pragma unroll
  for (int i = 0; i < 4; ++i) {
    int cl = threadIdx.y + 8 * i;
    dst[(size_t)(c0 + cl) * 1024 + hw0 + threadIdx.x] = tile[threadIdx.x][cl];
  }
}

// ---------------- K6: scalars ---------------------------------------------
__global__ __launch_bounds__(256) void k_finalize(const int* __restrict__ hist,
                                                  const float* __restrict__ total,
                                                  float* __restrict__ out_scalars) {
  __shared__ float s_nnz[256], s_ent[256];
  float nnz = 0.f, ent = 0.f;
  for (int i = threadIdx.x; i < KTOT; i += 256) {
    int h = hist[i];
    if (h > 0) {
      float p = (float)h * (1.0f / (float)NTOT);
      nnz += 1.0f;
      ent += p * logf(p + 1e-10f);
    }
  }
  s_nnz[threadIdx.x] = nnz; s_ent[threadIdx.x] = ent;
  __syncthreads();
  for (int stride = 128; stride >= 1; stride >>= 1) {
    if (threadIdx.x < stride) {
      s_nnz[threadIdx.x] += s_nnz[threadIdx.x + stride];
      s_ent[threadIdx.x] += s_ent[threadIdx.x + stride];
    }
    __syncthreads();
  }
  if (threadIdx.x == 0) {
    float tot = *total;
    out_scalars[0] = 1.25f * tot / ((float)NTOT * (float)DDIM);  // loss
    out_scalars[1] = tot / (float)NTOT;                          // quant_error
    out_scalars[2] = s_nnz[0] / (float)KTOT;                     // utilization
    out_scalars[3] = expf(-s_ent[0]);                            // perplexity
  }
}

extern "C" void kernel_launch(void* const* d_in, const int* in_sizes, int n_in,
                              void* d_out, int out_size, void* d_ws, size_t ws_size,
                              hipStream_t stream) {
  const float* z  = (const float*)d_in[0];   // [16,256,32,32]
  const float* cb = (const float*)d_in[1];   // [8192,256]
  float* out = (float*)d_out;                // 16*256*32*32 floats + 4 scalars

  char* ws = (char*)d_ws;
  float* zt     = (float*)(ws + 0);                    // 16,777,216 B
  float* cnorm  = (float*)(ws + 16777216);             //     32,768 B
  float* minval = (float*)(ws + 16809984);             //    131,072 B
  int*   minidx = (int*)  (ws + 16941056);             //    131,072 B
  int*   token  = (int*)  (ws + 17072128);             //     65,536 B
  int*   hist   = (int*)  (ws + 17137664);             //     32,768 B
  float* total  = (float*)(ws + 17170432);             //         16 B

  hipMemsetAsync(ws + 17137664, 0, 32768 + 16, stream);

  k_transpose_z<<<dim3(32, 8, 16), dim3(32, 8), 0, stream>>>(z, zt);
  k_cnorm<<<KTOT / 8, 256, 0, stream>>>(cb, cnorm);
  k_gemm_argmin<<<dim3(NTOT / 128, KSPLIT), 256, 0, stream>>>(zt, cb, cnorm,
                                                              minval, minidx);
  k_combine<<<NTOT / 256, 256, 0, stream>>>(minval, minidx, token);
  k_stats<<<NTOT, 256, 0, stream>>>(zt, cb, token, total, hist);
  k_write_out<<<dim3(32, 8, 16), dim3(32, 8), 0, stream>>>(cb, token, out);
  k_finalize<<<1, 256, 0, stream>>>(hist, total, out + (size_t)16 * 256 * 1024);
}